// GANSageEncoder_1090921693298
// MI455X (gfx1250) — compile-verified
//
#include <hip/hip_runtime.h>

// ---------------------------------------------------------------------------
// GraphSAGE 2-layer encoder for MI455X (gfx1250, wave32).
//  layer: out = (scatter_mean(x, src->dst)) @ Wl^T + b + x @ Wr^T   [+ relu]
// Matrix math in full fp32 via V_WMMA_F32_16X16X4_F32 (workload is scatter/
// bandwidth bound; fp32 WMMA keeps bit-compatible precision with reference).
// ---------------------------------------------------------------------------

typedef __attribute__((ext_vector_type(2))) float v2f;
typedef __attribute__((ext_vector_type(8))) float v8f;

__global__ void zero_f32(float* __restrict__ p, long long n) {
  long long i = (long long)blockIdx.x * blockDim.x + threadIdx.x;
  long long stride = (long long)gridDim.x * blockDim.x;
  for (; i < n; i += stride) p[i] = 0.0f;
}

__global__ void degree_kernel(const int* __restrict__ dst,
                              float* __restrict__ deg, int E) {
  int e = blockIdx.x * blockDim.x + threadIdx.x;
  if (e < E) atomicAdd(&deg[dst[e]], 1.0f);
}

// One thread handles 4 consecutive features of one edge: float4 gather from
// x[src], 4 global_atomic_add_f32 into agg[dst]. Accumulator (<=51MB) is
// L2-resident on the 192MB L2, so atomics resolve in-cache.
template <int D>
__global__ void scatter_add_kernel(const float* __restrict__ feat,
                                   const int* __restrict__ src,
                                   const int* __restrict__ dst,
                                   float* __restrict__ agg, int E) {
  constexpr int PER_EDGE = D / 4;
  long long tid = (long long)blockIdx.x * blockDim.x + threadIdx.x;
  long long total = (long long)E * PER_EDGE;
  if (tid >= total) return;
  int e = (int)(tid / PER_EDGE);
  int f = (int)(tid % PER_EDGE) * 4;
  int s = src[e], d = dst[e];
  float4 v = *(const float4*)(feat + (size_t)s * D + f);
  float* p = agg + (size_t)d * D + f;
  atomicAdd(p + 0, v.x);
  atomicAdd(p + 1, v.y);
  atomicAdd(p + 2, v.z);
  atomicAdd(p + 3, v.w);
}

// Fused SAGE linear: out[MxN] = (aggsum/deg) @ Wl^T + bias + xin @ Wr^T.
// One wave32 per 16x16 output tile, K stepped by 4 with fp32 WMMA.
// A-frag (16x4 f32): lanes 0-15 -> M=lane, {K=k0, k0+1}; lanes 16-31 ->
// M=lane-16, {K=k0+2, k0+3}. B-frag (4x16): symmetric over N columns.
// C/D: VGPR r holds M=r (lanes 0-15) / M=8+r (lanes 16-31), N=lane%16.
template <int K, bool RELU>
__launch_bounds__(256)
__global__ void sage_gemm_wmma(const float* __restrict__ aggsum,  // [M,K] raw sums
                               const float* __restrict__ deg,     // [M]
                               const float* __restrict__ xin,     // [M,K]
                               const float* __restrict__ Wl,      // [N,K]
                               const float* __restrict__ Wr,      // [N,K]
                               const float* __restrict__ bias,    // [N]
                               float* __restrict__ out,           // [M,N]
                               int M, int N) {
  const int lane = threadIdx.x & 31;
  const int wave = threadIdx.x >> 5;
  const int tile = blockIdx.x * (blockDim.x >> 5) + wave;
  const int tilesN = N >> 4;
  const int nTiles = (M >> 4) * tilesN;
  if (tile >= nTiles) return;  // wave-uniform: EXEC stays all-1s for WMMA

  const int tm = tile / tilesN;
  const int tn = tile - tm * tilesN;

  const int m  = lane & 15;   // A row within tile / B,C,D column within tile
  const int kh = lane >> 4;   // K-half select for A/B fragments

  const int arow = tm * 16 + m;
  const int bcol = tn * 16 + m;

  const float invd = 1.0f / fmaxf(deg[arow], 1.0f);  // PyG mean w/ deg>=1

  const float* aP = aggsum + (size_t)arow * K + 2 * kh;
  const float* xP = xin    + (size_t)arow * K + 2 * kh;
  const float* lP = Wl     + (size_t)bcol * K + 2 * kh;
  const float* rP = Wr     + (size_t)bcol * K + 2 * kh;

  v8f acc = {};

#pragma unroll 8
  for (int k0 = 0; k0 < K; k0 += 4) {   // (aggsum * invd) @ Wl^T
    v2f a, b;
    a.x = aP[k0] * invd;
    a.y = aP[k0 + 1] * invd;
    b.x = lP[k0];
    b.y = lP[k0 + 1];
    acc = __builtin_amdgcn_wmma_f32_16x16x4_f32(false, a, false, b,
                                                (short)0, acc, false, false);
  }

#pragma unroll 8
  for (int k0 = 0; k0 < K; k0 += 4) {   // xin @ Wr^T
    v2f a, b;
    a.x = xP[k0];
    a.y = xP[k0 + 1];
    b.x = rP[k0];
    b.y = rP[k0 + 1];
    acc = __builtin_amdgcn_wmma_f32_16x16x4_f32(false, a, false, b,
                                                (short)0, acc, false, false);
  }

  const float bv = bias[bcol];
#pragma unroll
  for (int r = 0; r < 8; ++r) {
    float v = acc[r] + bv;
    if (RELU) v = fmaxf(v, 0.0f);
    out[(size_t)(tm * 16 + r + 8 * kh) * N + bcol] = v;
  }
}

extern "C" void kernel_launch(void* const* d_in, const int* in_sizes, int n_in,
                              void* d_out, int out_size, void* d_ws, size_t ws_size,
                              hipStream_t stream) {
  (void)n_in; (void)out_size; (void)ws_size;

  const float* x   = (const float*)d_in[0];
  const int*   ei  = (const int*)  d_in[1];
  const float* W1l = (const float*)d_in[2];
  const float* b1  = (const float*)d_in[3];
  const float* W1r = (const float*)d_in[4];
  const float* W2l = (const float*)d_in[5];
  const float* b2  = (const float*)d_in[6];
  const float* W2r = (const float*)d_in[7];
  float* out = (float*)d_out;

  const int D1 = in_sizes[3];        // 256 (hidden)
  const int D2 = in_sizes[6];        // 128 (out)
  const int D0 = in_sizes[2] / D1;   // 128 (in)
  const int M  = in_sizes[0] / D0;   // 50000 nodes (= 3125 * 16, no tail)
  const int E  = in_sizes[1] / 2;    // 800000 edges
  const int* src = ei;               // edge_index[0]
  const int* dst = ei + E;           // edge_index[1]

  // Workspace layout (fp32): deg[M] | bufA[M*D1] (scatter acc, reused) | h[M*D1]
  float* deg  = (float*)d_ws;
  float* bufA = deg + M;
  float* h    = bufA + (size_t)M * D1;

  // ---- Layer 1 ----
  zero_f32<<<4096, 256, 0, stream>>>(deg, (long long)M * D0 + M);  // deg+bufA contiguous
  degree_kernel<<<(E + 255) / 256, 256, 0, stream>>>(dst, deg, E);
  {
    long long total = (long long)E * (128 / 4);
    scatter_add_kernel<128><<<(int)((total + 255) / 256), 256, 0, stream>>>(
        x, src, dst, bufA, E);
  }
  {
    int tiles = (M / 16) * (D1 / 16);
    sage_gemm_wmma<128, true><<<(tiles + 7) / 8, 256, 0, stream>>>(
        bufA, deg, x, W1l, W1r, b1, h, M, D1);
  }

  // ---- Layer 2 ----
  zero_f32<<<4096, 256, 0, stream>>>(bufA, (long long)M * D1);
  {
    long long total = (long long)E * (256 / 4);
    scatter_add_kernel<256><<<(int)((total + 255) / 256), 256, 0, stream>>>(
        h, src, dst, bufA, E);
  }
  {
    int tiles = (M / 16) * (D2 / 16);
    sage_gemm_wmma<256, false><<<(tiles + 7) / 8, 256, 0, stream>>>(
        bufA, deg, h, W2l, W2r, b2, out, M, D2);
  }
}